// AttentionHead_34660386079362
// MI455X (gfx1250) — compile-verified
//
#include <hip/hip_runtime.h>
#include <math.h>

#define IN_DIM   128
#define OUT_DIM  32
#define LEAKY    0.2f

typedef __attribute__((ext_vector_type(2)))  float     v2f;
typedef __attribute__((ext_vector_type(4)))  float     v4f;
typedef __attribute__((ext_vector_type(8)))  float     v8f;

// ---------------- helpers: order-preserving float<->uint for atomic max ----
__device__ __forceinline__ unsigned fenc(float f) {
    unsigned b = __float_as_uint(f);
    return (b & 0x80000000u) ? ~b : (b | 0x80000000u);
}
__device__ __forceinline__ float fdec(unsigned k) {
    return (k & 0x80000000u) ? __uint_as_float(k ^ 0x80000000u)
                             : __uint_as_float(~k);
}

// ---------------- K0: init scratch (vectorized) ----------------------------
__global__ void k_init(float* __restrict__ agg, float* __restrict__ denom,
                       unsigned* __restrict__ emax, int n_nodes) {
    int i = blockIdx.x * blockDim.x + threadIdx.x;
    int total4 = n_nodes * OUT_DIM / 4;
    if (i < total4) ((v4f*)agg)[i] = (v4f)0.0f;
    if (i < n_nodes) { denom[i] = 0.0f; emax[i] = fenc(-1e30f); }
}

// ---------------- K1: z = X @ W via WMMA (fp32-exact) ----------------------
// wave32; each wave owns MT m-tiles (16 rows each) x full 32 output cols.
// Out-of-range tiles are CLAMPED to the last tile: duplicated compute/stores
// write identical data to identical addresses, keeping control flow uniform
// and EXEC all-1s for every WMMA (no saveexec predication).
//
// W is staged in LDS k-pair interleaved: Wl[kp][n][p] = W[2*kp+p][n],
// row stride WSTR=48 pairs. A B-fragment {W[k'][n], W[k'+1][n]} is then one
// aligned ds_load_b64, and the half-lane kp offset (48*8B = 384B == 32 banks
// mod 64) puts lanes 16-31 on banks 32-63: conflict-free, no repack movs.
#define MT   4
#define WSTR 48   // padded row stride in k-pairs (>=32, chosen for bank split)
__global__ __launch_bounds__(128)
void k_gemm(const float* __restrict__ X, const float* __restrict__ W,
            float* __restrict__ Z, int n_tiles) {
    __shared__ float Wl[(IN_DIM / 2) * WSTR * 2];   // 24 KB
    int tid = threadIdx.x;
    for (int idx = tid; idx < IN_DIM * OUT_DIM; idx += 128) {
        int kk = idx / OUT_DIM, n = idx % OUT_DIM;
        Wl[(kk >> 1) * (WSTR * 2) + n * 2 + (kk & 1)] = W[idx];
    }
    __syncthreads();

    const int wave = tid >> 5;
    const int lane = tid & 31;
    const int half = lane >> 4;       // 0: lanes 0-15, 1: lanes 16-31
    const int ln   = lane & 15;

    const int tile0 = (blockIdx.x * 4 + wave) * MT;
    if (tile0 >= n_tiles) return;     // wave-uniform early out

    int mt[MT];
    const float* aptr[MT];
#pragma unroll
    for (int t = 0; t < MT; ++t) {
        int m = tile0 + t;
        mt[t] = (m < n_tiles) ? m : (n_tiles - 1);   // clamp (uniform data dup)
        aptr[t] = X + (size_t)(mt[t] * 16 + ln) * IN_DIM + 2 * half;
    }

    v8f acc0[MT], acc1[MT];
#pragma unroll
    for (int t = 0; t < MT; ++t) { acc0[t] = (v8f)0.0f; acc1[t] = (v8f)0.0f; }

#pragma unroll 2
    for (int k = 0; k < IN_DIM; k += 4) {
        // A frags: 4 independent b64 loads, issued before any wait
        v2f a[MT];
#pragma unroll
        for (int t = 0; t < MT; ++t)
            a[t] = *(const v2f*)(aptr[t] + k);
        // B frags: single ds_load_b64 each (pair-interleaved layout)
        const float* brow = Wl + ((k >> 1) + half) * (WSTR * 2);
        v2f b0 = *(const v2f*)(brow + ln * 2);
        v2f b1 = *(const v2f*)(brow + (16 + ln) * 2);
#pragma unroll
        for (int t = 0; t < MT; ++t) {
            acc0[t] = __builtin_amdgcn_wmma_f32_16x16x4_f32(
                false, a[t], false, b0, (short)0, acc0[t], false, false);
            acc1[t] = __builtin_amdgcn_wmma_f32_16x16x4_f32(
                false, a[t], false, b1, (short)0, acc1[t], false, false);
        }
    }

    // D layout: VGPR r holds row (16*tile + r + 8*half), col = ln (+16 for acc1)
#pragma unroll
    for (int t = 0; t < MT; ++t) {
        int rowbase = mt[t] * 16 + 8 * half;
#pragma unroll
        for (int r = 0; r < 8; ++r) {
            Z[(size_t)(rowbase + r) * OUT_DIM + ln]      = acc0[t][r];
            Z[(size_t)(rowbase + r) * OUT_DIM + 16 + ln] = acc1[t][r];
        }
    }
}

// GEMM tail for n_nodes % 16 rows (not hit for 100000, kept for safety)
__global__ void k_gemm_tail(const float* __restrict__ X, const float* __restrict__ W,
                            float* __restrict__ Z, int row0, int n_nodes) {
    int idx = blockIdx.x * blockDim.x + threadIdx.x;
    int nrows = n_nodes - row0;
    if (idx >= nrows * OUT_DIM) return;
    int r = row0 + idx / OUT_DIM, c = idx % OUT_DIM;
    float s = 0.0f;
    for (int k = 0; k < IN_DIM; ++k)
        s = fmaf(X[(size_t)r * IN_DIM + k], W[k * OUT_DIM + c], s);
    Z[(size_t)r * OUT_DIM + c] = s;
}

// ---------------- K2: per-node attention scalars ---------------------------
__global__ void k_scores(const float* __restrict__ Z,
                         const float* __restrict__ a_src,
                         const float* __restrict__ a_dst,
                         float* __restrict__ s_src, float* __restrict__ s_dst,
                         int n_nodes) {
    int n = blockIdx.x * blockDim.x + threadIdx.x;
    if (n >= n_nodes) return;
    const float* zp = Z + (size_t)n * OUT_DIM;
    float ss = 0.0f, sd = 0.0f;
#pragma unroll
    for (int j = 0; j < OUT_DIM; j += 4) {
        v4f z  = *(const v4f*)(zp + j);
        v4f as = *(const v4f*)(a_src + j);
        v4f ad = *(const v4f*)(a_dst + j);
        ss += z.x * as.x + z.y * as.y + z.z * as.z + z.w * as.w;
        sd += z.x * ad.x + z.y * ad.y + z.z * ad.z + z.w * ad.w;
    }
    s_src[n] = ss;
    s_dst[n] = sd;
}

// ---------------- K3: edge logits + segment max ----------------------------
__global__ void k_edge_max(const int* __restrict__ src, const int* __restrict__ dst,
                           const float* __restrict__ s_src, const float* __restrict__ s_dst,
                           float* __restrict__ e, unsigned* __restrict__ emax,
                           int n_edges) {
    int i = blockIdx.x * blockDim.x + threadIdx.x;
    if (i >= n_edges) return;
    int d = dst[i];
    float v = s_src[src[i]] + s_dst[d];
    v = v > 0.0f ? v : LEAKY * v;
    e[i] = v;
    atomicMax(&emax[d], fenc(v));
}

// ---------------- K4: exp + segment sum ------------------------------------
__global__ void k_edge_exp(const int* __restrict__ dst, float* __restrict__ e,
                           const unsigned* __restrict__ emax,
                           float* __restrict__ denom, int n_edges) {
    int i = blockIdx.x * blockDim.x + threadIdx.x;
    if (i >= n_edges) return;
    int d = dst[i];
    float ex = expf(e[i] - fdec(emax[d]));
    e[i] = ex;
    atomicAdd(&denom[d], ex);
}

// ---------------- K5: weighted aggregation (sorted dst, carry-save) --------
#define EPW 512   // edges per wave
#define PFD 8     // prefetch distance (edges)
__global__ __launch_bounds__(256)
void k_agg(const int* __restrict__ src, const int* __restrict__ dst,
           const float* __restrict__ e, const float* __restrict__ denom,
           const float* __restrict__ Z, float* __restrict__ agg, int n_edges) {
    int wave = blockIdx.x * (blockDim.x >> 5) + (threadIdx.x >> 5);
    int lane = threadIdx.x & 31;              // lane == output feature
    int e0 = wave * EPW;
    if (e0 >= n_edges) return;
    int e1 = e0 + EPW; if (e1 > n_edges) e1 = n_edges;
    int last = e1 - 1;

    float acc = 0.0f;
    int cur = dst[e0];
    // depth-1 software pipeline on the z[src] gather
    float zv = Z[(size_t)src[e0] * OUT_DIM + lane];
    for (int i = e0; i < e1; ++i) {
        float zcur = zv;
        int inext = (i + 1 < e1) ? (i + 1) : last;
        zv = Z[(size_t)src[inext] * OUT_DIM + lane];
        int ipf = (i + PFD < e1) ? (i + PFD) : last;
        __builtin_prefetch(&Z[(size_t)src[ipf] * OUT_DIM], 0, 0);

        int d = dst[i];                        // broadcast load
        if (d != cur) {                        // uniform branch (sorted dst)
            atomicAdd(&agg[(size_t)cur * OUT_DIM + lane], acc);
            acc = 0.0f;
            cur = d;
        }
        acc = fmaf(e[i] / denom[d], zcur, acc);
    }
    atomicAdd(&agg[(size_t)cur * OUT_DIM + lane], acc);
}

// ---------------- K6: ReLU writeback (vectorized) --------------------------
__global__ void k_relu(const float* __restrict__ agg, float* __restrict__ out,
                       int total4) {
    int i = blockIdx.x * blockDim.x + threadIdx.x;
    if (i >= total4) return;
    v4f v = ((const v4f*)agg)[i];
    v4f r;
    r.x = fmaxf(v.x, 0.0f); r.y = fmaxf(v.y, 0.0f);
    r.z = fmaxf(v.z, 0.0f); r.w = fmaxf(v.w, 0.0f);
    ((v4f*)out)[i] = r;
}

// ---------------- host launcher --------------------------------------------
extern "C" void kernel_launch(void* const* d_in, const int* in_sizes, int n_in,
                              void* d_out, int out_size, void* d_ws, size_t ws_size,
                              hipStream_t stream) {
    const float* X     = (const float*)d_in[0];   // [n, 128]
    const float* W     = (const float*)d_in[1];   // [128, 32]
    const float* a_src = (const float*)d_in[2];   // [32]
    const float* a_dst = (const float*)d_in[3];   // [32]
    const int*   src   = (const int*)d_in[4];     // [E]
    const int*   dst   = (const int*)d_in[5];     // [E]
    float*       out   = (float*)d_out;           // [n, 32]

    const int n_nodes = in_sizes[0] / IN_DIM;
    const int n_edges = in_sizes[4];

    // carve scratch
    char* p = (char*)d_ws;
    float*    Z     = (float*)p;    p += (size_t)n_nodes * OUT_DIM * sizeof(float);
    float*    agg   = (float*)p;    p += (size_t)n_nodes * OUT_DIM * sizeof(float);
    float*    e     = (float*)p;    p += (size_t)n_edges * sizeof(float);
    float*    ssrc  = (float*)p;    p += (size_t)n_nodes * sizeof(float);
    float*    sdst  = (float*)p;    p += (size_t)n_nodes * sizeof(float);
    unsigned* emax  = (unsigned*)p; p += (size_t)n_nodes * sizeof(unsigned);
    float*    denom = (float*)p;

    const int T = 256;
    // K0: init (grid must cover max(total/4, n_nodes) threads)
    {
        int total4 = n_nodes * OUT_DIM / 4;
        int span = total4 > n_nodes ? total4 : n_nodes;
        k_init<<<(span + T - 1) / T, T, 0, stream>>>(agg, denom, emax, n_nodes);
    }
    // K1: GEMM
    {
        int n_tiles = n_nodes / 16;
        if (n_tiles > 0) {
            int waves  = (n_tiles + MT - 1) / MT;
            int blocks = (waves + 3) / 4;
            k_gemm<<<blocks, 128, 0, stream>>>(X, W, Z, n_tiles);
        }
        int rem = n_nodes & 15;
        if (rem) {
            int total = rem * OUT_DIM;
            k_gemm_tail<<<(total + T - 1) / T, T, 0, stream>>>(X, W, Z,
                                                               n_nodes - rem, n_nodes);
        }
    }
    // K2: node scores
    k_scores<<<(n_nodes + T - 1) / T, T, 0, stream>>>(Z, a_src, a_dst,
                                                      ssrc, sdst, n_nodes);
    // K3: edge logits + max
    k_edge_max<<<(n_edges + T - 1) / T, T, 0, stream>>>(src, dst, ssrc, sdst,
                                                        e, emax, n_edges);
    // K4: exp + denom
    k_edge_exp<<<(n_edges + T - 1) / T, T, 0, stream>>>(dst, e, emax, denom, n_edges);
    // K5: aggregation
    {
        int waves  = (n_edges + EPW - 1) / EPW;
        int blocks = (waves + 7) / 8;     // 8 waves / 256-thread block
        k_agg<<<blocks, 256, 0, stream>>>(src, dst, e, denom, Z, agg, n_edges);
    }
    // K6: ReLU writeback
    {
        int total4 = n_nodes * OUT_DIM / 4;
        k_relu<<<(total4 + T - 1) / T, T, 0, stream>>>(agg, out, total4);
    }
}